// SPNNeuronVectorized_48275432407638
// MI455X (gfx1250) — compile-verified
//
#include <hip/hip_runtime.h>
#include <math.h>

// ---------------------------------------------------------------------------
// SPN neuron layer on MI455X (gfx1250, wave32).
// Roofline: 64MB of x @ 23.3 TB/s => ~2.8us floor; ~2 KFLOP/sample of VALU.
// Design:
//   * prep kernel folds means/stds/log-softmax/perm into quadratic coeffs:
//       lp(f,k) = A*x^2 + B*x + C,  Cw[g,k] = C(f0)+C(f1)+logw[g,k]
//   * main kernel streams x with the Tensor Data Mover (tensor_load_to_lds,
//     TENSORcnt double-buffering); TDM pad_interval/pad_amount gives a
//     129-dword LDS row stride (conflict-free column reads)
//   * 16 samples per wave-tile; wave halves split the 64 groups to match the
//     V_WMMA_F32_16X16X4_F32 A-operand layout (K chunks of 16 groups), so ONE
//     WMMA against a ones B-matrix performs the final 64-group reduction and
//     deposits all 16 per-sample totals in D column 0.
//   * LSE inner loop uses raw v_exp_f32/v_log_f32 (args provably in-range:
//     exp2 of <=0, log2 of [1,3]) with base conversion folded into FMAs.
// ---------------------------------------------------------------------------

#define BATCH   131072
#define NFEAT   128
#define MULT    3
#define NGROUP  64
#define ROWS    16            // samples per tile (wave halves share samples)
#define ROW_DW  129           // padded LDS row stride (dwords), via TDM padding
#define TILE_DW (ROWS * ROW_DW)
#define WPB     4             // waves per block
#define THREADS (WPB * 32)
#define NBLOCKS 512

typedef unsigned int u32;
typedef __attribute__((ext_vector_type(4))) u32   u32x4;
typedef __attribute__((ext_vector_type(8))) int   i32x8;
typedef __attribute__((ext_vector_type(4))) int   i32x4;
typedef __attribute__((ext_vector_type(2))) float v2f;
typedef __attribute__((ext_vector_type(8))) float v8f;

#if defined(__has_builtin)
#if __has_builtin(__builtin_amdgcn_tensor_load_to_lds) && \
    __has_builtin(__builtin_amdgcn_s_wait_tensorcnt)
#define HAVE_TDM 1
#endif
#if __has_builtin(__builtin_amdgcn_wmma_f32_16x16x4_f32)
#define HAVE_WMMA4 1
#endif
#if __has_builtin(__builtin_amdgcn_exp2f) && __has_builtin(__builtin_amdgcn_logf)
#define FAST_EXP2(a) __builtin_amdgcn_exp2f(a)   // v_exp_f32
#define FAST_LOG2(a) __builtin_amdgcn_logf(a)    // v_log_f32 (base-2)
#else
#define FAST_EXP2(a) exp2f(a)
#define FAST_LOG2(a) log2f(a)
#endif
#endif

#define LOG2E_F 1.44269504088896340736f
#define LN2_F   0.69314718055994530942f

// ---------------------------------------------------------------------------
// Prep: coef[g][16] = {A0,B0,A1,B1}[k=0..2], {Cw0,Cw1,Cw2, packed(f0,f1)}
// ---------------------------------------------------------------------------
__global__ void spn_prep(const float* __restrict__ means,
                         const float* __restrict__ stds,
                         const float* __restrict__ sw,
                         const int*   __restrict__ perm,
                         float* __restrict__ coef)
{
    int g = threadIdx.x;
    if (g >= NGROUP) return;
    int f0 = perm[2 * g], f1 = perm[2 * g + 1];

    float w0 = sw[g * 3 + 0], w1 = sw[g * 3 + 1], w2 = sw[g * 3 + 2];
    float wm = fmaxf(w0, fmaxf(w1, w2));
    float lse = wm + __logf(__expf(w0 - wm) + __expf(w1 - wm) + __expf(w2 - wm));

    const float HLOG2PI = 0.91893853320467274178f; // 0.5*ln(2*pi)
#pragma unroll
    for (int k = 0; k < MULT; ++k) {
        float mu0 = means[f0 * MULT + k], s0 = stds[f0 * MULT + k];
        float mu1 = means[f1 * MULT + k], s1 = stds[f1 * MULT + k];
        float i0 = 1.0f / (s0 * s0), i1 = 1.0f / (s1 * s1);
        coef[g * 16 + 4 * k + 0] = -0.5f * i0;
        coef[g * 16 + 4 * k + 1] = mu0 * i0;
        coef[g * 16 + 4 * k + 2] = -0.5f * i1;
        coef[g * 16 + 4 * k + 3] = mu1 * i1;
        float C0 = -0.5f * mu0 * mu0 * i0 - __logf(s0) - HLOG2PI;
        float C1 = -0.5f * mu1 * mu1 * i1 - __logf(s1) - HLOG2PI;
        coef[g * 16 + 12 + k] = C0 + C1 + (sw[g * 3 + k] - lse);
    }
    coef[g * 16 + 15] = __int_as_float(f0 | (f1 << 16));
}

// ---------------------------------------------------------------------------
// TDM tile fetch: ROWS x 128 f32, global -> LDS (async, TENSORcnt).
// D# packed per CDNA5 ISA ch.8; pad_interval=128dw + pad_amount=1dw => 129dw
// row stride in LDS.
// ---------------------------------------------------------------------------
#ifdef HAVE_TDM
__device__ __forceinline__ void tile_fetch(const float* src, float* dst)
{
    const unsigned long long ga = (unsigned long long)src;
    const u32 ldsa = (u32)(unsigned long long)dst;   // low 32 bits = LDS offset
    u32x4 g0 = { 1u,                                  // count=1, user descriptor
                 ldsa,                                 // lds_addr      [63:32]
                 (u32)ga,                              // global_addr lo
                 ((u32)(ga >> 32) & 0x01FFFFFFu) | (2u << 30) }; // ga hi + type=2
    i32x8 g1 = {
        (2 << 16) | (1 << 20) | (6 << 22),  // data_size=4B, pad_enable, interval=128dw, amount=1dw
        (int)(128u << 16),                  // tensor_dim0 = 128 (bits 79:48)
        0,                                  // tensor_dim1 lo16 = 0 (131072 & 0xffff)
        2 | (128 << 16),                    // tensor_dim1 hi16 = 2, tile_dim0 = 128
        ROWS,                               // tile_dim1 = 16, tile_dim2 = 0
        128,                                // tensor_dim0_stride = 128 elements
        0, 0
    };
    i32x4 z4 = { 0, 0, 0, 0 };
#if __clang_major__ >= 23
    i32x8 z8 = { 0, 0, 0, 0, 0, 0, 0, 0 };
    __builtin_amdgcn_tensor_load_to_lds(g0, g1, z4, z4, z8, 0);
#else
    __builtin_amdgcn_tensor_load_to_lds(g0, g1, z4, z4, 0);
#endif
}
#define TILE_WAIT(n) __builtin_amdgcn_s_wait_tensorcnt(n)
#else
__device__ __forceinline__ void tile_fetch(const float* src, float* dst)
{
    const int lane = threadIdx.x & 31;
#pragma unroll
    for (int r = 0; r < ROWS; ++r)
#pragma unroll
        for (int c = 0; c < NFEAT; c += 32)
            dst[r * ROW_DW + c + lane] = src[r * NFEAT + c + lane];
}
#define TILE_WAIT(n) ((void)0)
#endif

// One product node: 3-component weighted logsumexp of pair-sum log-probs.
// v_k - m <= 0 so raw v_exp_f32 is safe; s in [1,3] so raw v_log_f32 is safe.
__device__ __forceinline__ float lse_group(const float* __restrict__ c,
                                           const float* __restrict__ row)
{
    const float4 c0 = ((const float4*)c)[0];
    const float4 c1 = ((const float4*)c)[1];
    const float4 c2 = ((const float4*)c)[2];
    const float4 cw = ((const float4*)c)[3];
    const int   fp = __float_as_int(cw.w);
    const float x0 = row[fp & 0xFFFF];
    const float x1 = row[fp >> 16];
    const float q0 = x0 * x0, q1 = x1 * x1;
    float v0 = fmaf(c0.x, q0, fmaf(c0.y, x0, fmaf(c0.z, q1, fmaf(c0.w, x1, cw.x))));
    float v1 = fmaf(c1.x, q0, fmaf(c1.y, x0, fmaf(c1.z, q1, fmaf(c1.w, x1, cw.y))));
    float v2 = fmaf(c2.x, q0, fmaf(c2.y, x0, fmaf(c2.z, q1, fmaf(c2.w, x1, cw.z))));
    float m  = fmaxf(v0, fmaxf(v1, v2));          // v_max3_num_f32
    float mb = m * LOG2E_F;
    float s  = FAST_EXP2(fmaf(v0, LOG2E_F, -mb))
             + FAST_EXP2(fmaf(v1, LOG2E_F, -mb))
             + FAST_EXP2(fmaf(v2, LOG2E_F, -mb));
    return fmaf(LN2_F, FAST_LOG2(s), m);
}

// ---------------------------------------------------------------------------
// Main kernel.
// ---------------------------------------------------------------------------
__global__ __launch_bounds__(THREADS) void spn_main(
    const float* __restrict__ x, const float* __restrict__ coef,
    float* __restrict__ out, int nTiles)
{
    __shared__ float tiles[WPB * 2 * TILE_DW];   // 66 KB: double-buffered tiles
    __shared__ float scoef[NGROUP * 16];         // 4 KB: coefficient table

    for (int i = threadIdx.x; i < NGROUP * 16; i += THREADS)
        scoef[i] = coef[i];
    __syncthreads();

    const int lane  = threadIdx.x & 31;
    const int wave  = __builtin_amdgcn_readfirstlane((int)(threadIdx.x >> 5));
    const int gwave = blockIdx.x * WPB + wave;
    const int nWaves = gridDim.x * WPB;

    float* buf0 = &tiles[wave * 2 * TILE_DW];
    float* buf1 = buf0 + TILE_DW;

    int t = gwave;
    if (t < nTiles) tile_fetch(x + (size_t)t * (ROWS * NFEAT), buf0);

    int parity = 0;
    for (; t < nTiles; t += nWaves, parity ^= 1) {
        const int nt = t + nWaves;
        if (nt < nTiles) {
            tile_fetch(x + (size_t)nt * (ROWS * NFEAT), parity ? buf0 : buf1);
            TILE_WAIT(1);                 // current tile complete
        } else {
            TILE_WAIT(0);                 // drain
        }
        asm volatile("" ::: "memory");    // TDM wrote LDS behind compiler's back

        // Wave-half group split matching WMMA 16x16x4 f32 A-operand layout:
        //   lanes 0-15 : sample=lane,    K=0 -> groups  0-15, K=1 -> 16-31
        //   lanes 16-31: sample=lane-16, K=2 -> groups 32-47, K=3 -> 48-63
        const float* row   = (parity ? buf1 : buf0) + (lane & 15) * ROW_DW;
        const float* cbase = scoef + ((lane >> 4) << 5) * 16;

        float accx = 0.0f, accy = 0.0f;
#pragma unroll 2
        for (int i = 0; i < 16; ++i)
            accx += lse_group(cbase + i * 16, row);
#pragma unroll 2
        for (int i = 0; i < 16; ++i)
            accy += lse_group(cbase + (16 + i) * 16, row);

#ifdef HAVE_WMMA4
        // totals[i] = sum_k A[i,k] : one WMMA against ones; EXEC is all 1s here.
        v2f a = { accx, accy };
        v2f b = { 1.0f, 1.0f };
        v8f c = {};
        v8f d = __builtin_amdgcn_wmma_f32_16x16x4_f32(
            /*neg_a=*/false, a, /*neg_b=*/false, b,
            /*c_mod=*/(short)0, c, /*reuse_a=*/false, /*reuse_b=*/false);
        // D column 0: lane 0 holds samples 0-7 (VGPR 0-7), lane 16 holds 8-15.
        if ((lane & 15) == 0) {
            float* o = out + (size_t)t * ROWS + (lane >> 4) * 8;
            ((float4*)o)[0] = make_float4(d[0], d[1], d[2], d[3]);
            ((float4*)o)[1] = make_float4(d[4], d[5], d[6], d[7]);
        }
#else
        // Fallback: combine the two wave halves with a lane-16 xor shuffle.
        float part = accx + accy;
        float total = part + __shfl_xor(part, 16, 32);
        if (lane < 16) out[(size_t)t * ROWS + lane] = total;
#endif
    }
}

// ---------------------------------------------------------------------------
extern "C" void kernel_launch(void* const* d_in, const int* in_sizes, int n_in,
                              void* d_out, int out_size, void* d_ws, size_t ws_size,
                              hipStream_t stream)
{
    const float* x     = (const float*)d_in[0];
    const float* means = (const float*)d_in[1];
    const float* stds  = (const float*)d_in[2];
    const float* sw    = (const float*)d_in[3];
    const int*   perm  = (const int*)d_in[4];
    float* out  = (float*)d_out;
    float* coef = (float*)d_ws;   // 64*16*4 = 4KB scratch

    spn_prep<<<1, 64, 0, stream>>>(means, stds, sw, perm, coef);

    const int nTiles = BATCH / ROWS; // 8192
    spn_main<<<NBLOCKS, THREADS, 0, stream>>>(x, coef, out, nTiles);
}